// ETLayer_16312285790825
// MI455X (gfx1250) — compile-verified
//
#include <hip/hip_runtime.h>
#include <math.h>

// ============================================================================
// ETLayer (equivariant transformer layer) for MI455X / gfx1250, fp32 + WMMA.
// B=48, H=256, T=512, TE=256.  All heavy GEMMs use V_WMMA_F32_16X16X4_F32.
// Workspace requirement: ~178 MB (44,326,912 floats).
//
// GEMM design: one wave owns a 16x64 C tile (4 x v8f accumulators), 8 waves
// per block -> 128x64 macro-tile.  All M are multiples of 128 (emb table is
// padded 1023->1024 rows; the pad row is never read since r in [0,1022]).
// The K-loop is branch-free and software-pipelined one k-step ahead: next
// iteration's 1 A-fragment + 4 B-fragments load while 4 WMMAs execute.
// 32-bit offsets from uniform bases -> SGPR-base + VGPR-offset addressing.
// ============================================================================

typedef float v2f __attribute__((ext_vector_type(2)));
typedef float v8f __attribute__((ext_vector_type(8)));

#define BB 48
#define HH 256
#define TT 512
#define TE 256
#define RR 1023   // 2T-1 (valid rows of emb_rel)
#define RP 1024   // padded rows

#define WMMA_F32(a, b, c) \
    __builtin_amdgcn_wmma_f32_16x16x4_f32(false, (a), false, (b), (short)0, (c), false, false)

// ---------------------------------------------------------------------------
// Batched fp32 WMMA GEMM:  C = act(A[M,K] @ op(B) + bias)
//   TRANSB=0: B is [K,N] row-major.  TRANSB=1: B is [N,K] row-major (C=A@B^T).
//   ACT=1: SiLU epilogue.
// Requires: M % 128 == 0, N % 64 == 0, K % 4 == 0.
// Fragment layouts per ISA 7.12.2 (32-bit A 16x4; B/C striped across lanes).
// ---------------------------------------------------------------------------
template <int TRANSB, int ACT>
__global__ void gemm_f32_wmma(const float* __restrict__ A,
                              const float* __restrict__ Bm,
                              const float* __restrict__ bias,
                              float* __restrict__ C,
                              int K, int lda, int ldb, int ldc,
                              long strideA, long strideB, long strideC)
{
    const int wave = threadIdx.x >> 5;
    const int lane = threadIdx.x & 31;
    const int m0 = (blockIdx.y * 8 + wave) * 16;
    const int n0 = blockIdx.x * 64;

    const float* __restrict__ Ab = A  + (long)blockIdx.z * strideA;
    const float* __restrict__ Bb = Bm + (long)blockIdx.z * strideB;
    float* __restrict__ Cb       = C  + (long)blockIdx.z * strideC;

    const int lr = lane & 15;        // row-in-tile (A) / col-in-tile (B,C)
    const int kh = (lane >> 4) * 2;  // which K pair this half-wave holds (0|2)

    // 32-bit lane offsets from uniform bases (all buffers < 2^31 elements)
    int aoff = (m0 + lr) * lda + kh;
    int boff, bstep, brow, btile;
    if (TRANSB) {  // b = { B[col, kk+kh], B[col, kk+kh+1] }
        boff = (n0 + lr) * ldb + kh;  bstep = 4;        brow = 1;    btile = 16 * ldb;
    } else {       // b = { B[kk+kh, col], B[kk+kh+1, col] }
        boff = kh * ldb + n0 + lr;    bstep = 4 * ldb;  brow = ldb;  btile = 16;
    }

    v8f acc[4] = {};

    // prologue: fragments for kk = 0
    v2f a  = {Ab[aoff],             Ab[aoff + 1]};
    v2f b0 = {Bb[boff],             Bb[boff + brow]};
    v2f b1 = {Bb[boff +     btile], Bb[boff +     btile + brow]};
    v2f b2 = {Bb[boff + 2 * btile], Bb[boff + 2 * btile + brow]};
    v2f b3 = {Bb[boff + 3 * btile], Bb[boff + 3 * btile + brow]};

    for (int kk = 4; kk < K; kk += 4) {
        aoff += 4; boff += bstep;
        // issue next k-step's loads while current WMMAs execute
        v2f an  = {Ab[aoff],             Ab[aoff + 1]};
        v2f bn0 = {Bb[boff],             Bb[boff + brow]};
        v2f bn1 = {Bb[boff +     btile], Bb[boff +     btile + brow]};
        v2f bn2 = {Bb[boff + 2 * btile], Bb[boff + 2 * btile + brow]};
        v2f bn3 = {Bb[boff + 3 * btile], Bb[boff + 3 * btile + brow]};

        acc[0] = WMMA_F32(a, b0, acc[0]);
        acc[1] = WMMA_F32(a, b1, acc[1]);
        acc[2] = WMMA_F32(a, b2, acc[2]);
        acc[3] = WMMA_F32(a, b3, acc[3]);

        a = an; b0 = bn0; b1 = bn1; b2 = bn2; b3 = bn3;
    }
    acc[0] = WMMA_F32(a, b0, acc[0]);
    acc[1] = WMMA_F32(a, b1, acc[1]);
    acc[2] = WMMA_F32(a, b2, acc[2]);
    acc[3] = WMMA_F32(a, b3, acc[3]);

    // C/D layout: VGPR r -> row m0+r (lanes 0-15) / m0+8+r (lanes 16-31)
    const int rbase = m0 + ((lane >> 4) << 3);
    #pragma unroll
    for (int nt = 0; nt < 4; ++nt) {
        const int col = n0 + nt * 16 + lr;
        const float bv = bias ? bias[col] : 0.f;
        #pragma unroll
        for (int r = 0; r < 8; ++r) {
            float v = acc[nt][r] + bv;
            if (ACT) v = v / (1.f + __expf(-v));  // SiLU
            Cb[(rbase + r) * ldc + col] = v;
        }
    }
}

// ---------------------------------------------------------------------------
// Timestep embedding table te[1024,256] (row 1023 is padding, never consumed)
// ---------------------------------------------------------------------------
__global__ void te_embed(float* __restrict__ te)
{
    const int i = blockIdx.x * blockDim.x + threadIdx.x;
    if (i >= RP * TE) return;
    const int row = i / TE, j = i % TE;
    const float pos = (float)(row - (TT - 1));
    const float c = -9.210340371976184f / (float)(TE / 2 - 1);  // -ln(1e4)/127
    if (j < TE / 2) te[i] = __sinf(pos * __expf(c * (float)j));
    else            te[i] = __cosf(pos * __expf(c * (float)(j - TE / 2)));
}

// ---------------------------------------------------------------------------
// merge_time: hm[(b*T+t), c] = h[b, c, t]
// ---------------------------------------------------------------------------
__global__ void merge_time(const float* __restrict__ h, float* __restrict__ hm)
{
    const long i = (long)blockIdx.x * blockDim.x + threadIdx.x;
    if (i >= (long)BB * HH * TT) return;
    const int t = (int)(i % TT);
    const int c = (int)((i / TT) % HH);
    const long b = i / ((long)TT * HH);
    hm[(b * TT + t) * HH + c] = h[i];
}

// ---------------------------------------------------------------------------
// GEMV: vx[row] = hidden[row,:] . xW2 + xb2   (one block per row)
// ---------------------------------------------------------------------------
__global__ void gemv_vx(const float* __restrict__ A, const float* __restrict__ w,
                        const float* __restrict__ bias, float* __restrict__ out)
{
    __shared__ float red[256];
    const int row = blockIdx.x;
    const int tid = threadIdx.x;
    red[tid] = A[(long)row * HH + tid] * w[tid];
    __syncthreads();
    for (int s = 128; s > 0; s >>= 1) {
        if (tid < s) red[tid] += red[tid + s];
        __syncthreads();
    }
    if (tid == 0) out[row] = red[0] + bias[0];
}

// ---------------------------------------------------------------------------
// Softmax with fused relative-position logit bias.
// logits[b,t,s] += q[b,t,:] . emb_rel[t-s+511,:], then row softmax (in place).
// One block per (b,t); 256 threads handle 2 s-values each.
// ---------------------------------------------------------------------------
__global__ void attn_softmax(const float* __restrict__ q,
                             const float* __restrict__ E,
                             float* __restrict__ logits)
{
    __shared__ float qrow[HH];
    __shared__ float red[256];
    const int t = blockIdx.x, b = blockIdx.y, tid = threadIdx.x;
    const float* qp = q + ((long)(b * TT + t)) * HH;
    qrow[tid] = qp[tid];
    __syncthreads();

    float* lp = logits + ((long)(b * TT + t)) * TT;
    float l[2];
    #pragma unroll
    for (int u = 0; u < 2; ++u) {
        const int s = tid + u * 256;
        const int r = t - s + (TT - 1);  // in [0, 1022]
        const float* Er = E + (long)r * HH;
        float dot = 0.f;
        #pragma unroll 4
        for (int j = 0; j < HH; ++j) dot += qrow[j] * Er[j];
        l[u] = lp[s] + dot;
    }

    float m = fmaxf(l[0], l[1]);
    red[tid] = m; __syncthreads();
    for (int s = 128; s > 0; s >>= 1) {
        if (tid < s) red[tid] = fmaxf(red[tid], red[tid + s]);
        __syncthreads();
    }
    m = red[0]; __syncthreads();

    const float e0 = __expf(l[0] - m), e1 = __expf(l[1] - m);
    red[tid] = e0 + e1; __syncthreads();
    for (int s = 128; s > 0; s >>= 1) {
        if (tid < s) red[tid] += red[tid + s];
        __syncthreads();
    }
    const float inv = 1.f / red[0];
    lp[tid] = e0 * inv;
    lp[tid + 256] = e1 * inv;
}

// ---------------------------------------------------------------------------
// bias_h: attn_out[b,t,h] += sum_s alpha[b,t,s] * emb_rel[t-s+511, h]
// One block per (b,t); thread = h; alpha row broadcast through LDS.
// ---------------------------------------------------------------------------
__global__ void bias_h_add(const float* __restrict__ alpha,
                           const float* __restrict__ E,
                           float* __restrict__ attn_out)
{
    __shared__ float arow[256];
    const int t = blockIdx.x, b = blockIdx.y, tid = threadIdx.x;
    const float* ap = alpha + ((long)(b * TT + t)) * TT;
    float acc = 0.f;
    for (int chunk = 0; chunk < TT; chunk += 256) {
        arow[tid] = ap[chunk + tid];
        __syncthreads();
        #pragma unroll 4
        for (int j = 0; j < 256; ++j) {
            const int r = t - (chunk + j) + (TT - 1);
            acc += arow[j] * E[r * HH + tid];
        }
        __syncthreads();
    }
    attn_out[((long)(b * TT + t)) * HH + tid] += acc;
}

// ---------------------------------------------------------------------------
// h_new[b,c,t] = h[b,c,t] + attn_out[b,t,c]
// ---------------------------------------------------------------------------
__global__ void write_h(const float* __restrict__ h,
                        const float* __restrict__ attn_out,
                        float* __restrict__ hout)
{
    const long i = (long)blockIdx.x * blockDim.x + threadIdx.x;
    if (i >= (long)BB * HH * TT) return;
    const int t = (int)(i % TT);
    const int c = (int)((i / TT) % HH);
    const long b = i / ((long)TT * HH);
    hout[i] = h[i] + attn_out[(b * TT + t) * HH + c];
}

// ---------------------------------------------------------------------------
// Equivariant coordinate update.  One block per (b,t):
//   w[s] = alpha[b,t,s]*vx[b,s];  sw = sum w;  sx[c] = sum w*x[b,c,s]
//   x_new[b,c,t] = x + x*sw - sx[c]
// ---------------------------------------------------------------------------
__global__ void coord_update(const float* __restrict__ alpha,
                             const float* __restrict__ vx,
                             const float* __restrict__ x,
                             float* __restrict__ xout)
{
    __shared__ float red[256];
    const int t = blockIdx.x, b = blockIdx.y, tid = threadIdx.x;
    const float* ap = alpha + ((long)(b * TT + t)) * TT;
    const float* vp = vx + (long)b * TT;
    const float* xb = x + (long)b * 3 * TT;

    float sw = 0.f, s0 = 0.f, s1 = 0.f, s2 = 0.f;
    #pragma unroll
    for (int u = 0; u < 2; ++u) {
        const int s = tid + u * 256;
        const float w = ap[s] * vp[s];
        sw += w;
        s0 += w * xb[s];
        s1 += w * xb[TT + s];
        s2 += w * xb[2 * TT + s];
    }

    float vals[4] = {sw, s0, s1, s2};
    float tot[4];
    #pragma unroll
    for (int k = 0; k < 4; ++k) {
        red[tid] = vals[k]; __syncthreads();
        for (int s = 128; s > 0; s >>= 1) {
            if (tid < s) red[tid] += red[tid + s];
            __syncthreads();
        }
        tot[k] = red[0]; __syncthreads();
    }

    if (tid == 0) {
        #pragma unroll
        for (int c = 0; c < 3; ++c) {
            const float xv = xb[c * TT + t];
            xout[((long)b * 3 + c) * TT + t] = xv + xv * tot[0] - tot[1 + c];
        }
    }
}

// ===========================================================================
extern "C" void kernel_launch(void* const* d_in, const int* in_sizes, int n_in,
                              void* d_out, int out_size, void* d_ws, size_t ws_size,
                              hipStream_t stream)
{
    (void)in_sizes; (void)n_in; (void)out_size; (void)ws_size;

    const float* x   = (const float*)d_in[0];
    const float* h   = (const float*)d_in[1];
    const float* qW1 = (const float*)d_in[2];
    const float* qb1 = (const float*)d_in[3];
    const float* qW2 = (const float*)d_in[4];
    const float* qb2 = (const float*)d_in[5];
    const float* kW1 = (const float*)d_in[6];
    const float* kb1 = (const float*)d_in[7];
    const float* kW2 = (const float*)d_in[8];
    const float* kb2 = (const float*)d_in[9];
    const float* vW1 = (const float*)d_in[10];
    const float* vb1 = (const float*)d_in[11];
    const float* vW2 = (const float*)d_in[12];
    const float* vb2 = (const float*)d_in[13];
    const float* xW1 = (const float*)d_in[14];
    const float* xb1 = (const float*)d_in[15];
    const float* xW2 = (const float*)d_in[16];
    const float* xb2 = (const float*)d_in[17];
    const float* teW = (const float*)d_in[18];
    const float* teb = (const float*)d_in[19];

    // ---- workspace layout (floats) -- total 44,326,912 floats ~ 178 MB ----
    float* ws   = (float*)d_ws;
    float* emb  = ws;                  // 1024*256 (padded)
    float* tmp  = emb + 262144;        // 24576*256 : te table / MLP hidden
    float* hm   = tmp + 6291456;       // 24576*256 : merged h; later attn_out
    float* qd   = hm  + 6291456;       // 24576*256
    float* kd   = qd  + 6291456;       // 24576*256
    float* vd   = kd  + 6291456;       // 24576*256
    float* vxd  = vd  + 6291456;       // 24576
    float* alph = vxd + 24576;         // 48*512*512
    float* attn = hm;                  // reuse hm region after MLPs

    float* xout = (float*)d_out;             // [48,3,512]
    float* hout = xout + (long)BB * 3 * TT;  // [48,256,512]

    const long M_MLP = (long)BB * TT;        // 24576
    const dim3 blk(256);

    // 1) timestep embedding table -> tmp (1024 rows), emb_rel = te @ teW + teb
    te_embed<<<(RP * TE + 255) / 256, blk, 0, stream>>>(tmp);
    gemm_f32_wmma<0, 0><<<dim3(HH / 64, RP / 128, 1), blk, 0, stream>>>(
        tmp, teW, teb, emb, TE, TE, HH, HH, 0, 0, 0);

    // 2) merge time dim: hm[bT, H]
    merge_time<<<(BB * HH * TT + 255) / 256, blk, 0, stream>>>(h, hm);

    const dim3 gMLP(HH / 64, (unsigned)(M_MLP / 128), 1);
    // 3) q = (silu(hm@qW1+qb1))@qW2+qb2
    gemm_f32_wmma<0, 1><<<gMLP, blk, 0, stream>>>(hm, qW1, qb1, tmp, HH, HH, HH, HH, 0, 0, 0);
    gemm_f32_wmma<0, 0><<<gMLP, blk, 0, stream>>>(tmp, qW2, qb2, qd,  HH, HH, HH, HH, 0, 0, 0);
    // 4) k
    gemm_f32_wmma<0, 1><<<gMLP, blk, 0, stream>>>(hm, kW1, kb1, tmp, HH, HH, HH, HH, 0, 0, 0);
    gemm_f32_wmma<0, 0><<<gMLP, blk, 0, stream>>>(tmp, kW2, kb2, kd,  HH, HH, HH, HH, 0, 0, 0);
    // 5) v
    gemm_f32_wmma<0, 1><<<gMLP, blk, 0, stream>>>(hm, vW1, vb1, tmp, HH, HH, HH, HH, 0, 0, 0);
    gemm_f32_wmma<0, 0><<<gMLP, blk, 0, stream>>>(tmp, vW2, vb2, vd,  HH, HH, HH, HH, 0, 0, 0);
    // 6) v_x = (silu(v@xW1+xb1))@xW2+xb2   (GEMV for the 256->1 layer)
    gemm_f32_wmma<0, 1><<<gMLP, blk, 0, stream>>>(vd, xW1, xb1, tmp, HH, HH, HH, HH, 0, 0, 0);
    gemv_vx<<<(unsigned)M_MLP, blk, 0, stream>>>(tmp, xW2, xb2, vxd);

    // 7) base logits: alpha[b] = q_b @ k_b^T   (batched NT)
    gemm_f32_wmma<1, 0><<<dim3(TT / 64, TT / 128, BB), blk, 0, stream>>>(
        qd, kd, nullptr, alph, HH, HH, HH, TT,
        (long)TT * HH, (long)TT * HH, (long)TT * TT);

    // 8) fused relative-position bias + row softmax (in place on alph)
    attn_softmax<<<dim3(TT, BB), blk, 0, stream>>>(qd, emb, alph);

    // 9) attn[b] = alpha_b @ v_b   (batched NN)
    gemm_f32_wmma<0, 0><<<dim3(HH / 64, TT / 128, BB), blk, 0, stream>>>(
        alph, vd, nullptr, attn, TT, TT, HH, HH,
        (long)TT * TT, (long)TT * HH, (long)TT * HH);

    // 10) attn += sum_s alpha * emb_rel[t-s+511]
    bias_h_add<<<dim3(TT, BB), blk, 0, stream>>>(alph, emb, attn);

    // 11) h_new = h + attn (transposed back to [B,H,T])
    write_h<<<(BB * HH * TT + 255) / 256, blk, 0, stream>>>(h, attn, hout);

    // 12) equivariant coordinate update -> x_new
    coord_update<<<dim3(TT, BB), blk, 0, stream>>>(alph, vxd, x, xout);
}